// CapsuleLayer_55224689492491
// MI455X (gfx1250) — compile-verified
//
#include <hip/hip_runtime.h>
#include <hip/hip_bf16.h>

// ---- problem constants (from reference) ----
#define B_   64
#define IN_  2048
#define DIN_ 16
#define J_   32
#define D_   32
#define JD_  1024          // J_*D_
#define SUH  1028          // padded LDS stride for u_hat / s_acc rows (bank-conflict pad)
#define SVJ  33            // padded per-j stride for staged v
#define SV   (32*SVJ)      // per-b stride for staged v
#define ICHUNK 16          // i values per workgroup

typedef __attribute__((ext_vector_type(2))) float v2f;
typedef __attribute__((ext_vector_type(8))) float v8f;

// LDS layout (floats): uhat[16*SUH] | sacc[16*SUH] | vls[16*SV] | cls[16*32]
#define LDS_FLOATS (16*SUH + 16*SUH + 16*SV + 16*32)

// Fused routing iteration:
//  recompute u_hat tile [16 b][1024 jd] per i via WMMA f32 16x16x4,
//  (iter>0) agreement + logit update + softmax over j,
//  accumulate s partial sums, atomically reduce into global s.
__global__ __launch_bounds__(256, 1)
void caps_route(const float* __restrict__ inp,   // [B, In, Din]
                const float* __restrict__ W,     // [J, In, D, Din]
                const float* __restrict__ vprev, // [B, J, D] (iter>0)
                float* __restrict__ blog,        // [B, J, In] logits
                float* __restrict__ sout,        // [B, J, D] accumulator (pre-zeroed)
                int iter) {
    extern __shared__ float smem[];
    float* uhat = smem;               // [16][SUH]
    float* sacc = uhat + 16*SUH;      // [16][SUH]
    float* vls  = sacc + 16*SUH;      // [16][SV]
    float* cls  = vls  + 16*SV;       // [16][32]

    const int tid  = threadIdx.x;
    const int b0   = blockIdx.y * 16;       // batch tile base
    const int i0   = blockIdx.x * ICHUNK;   // input-capsule chunk base
    const int wave = tid >> 5;
    const int lane = tid & 31;
    const int lo   = lane & 15;             // M/N index within 16
    const int hi   = lane >> 4;             // half-wave selector

    // zero partial-sum tile
    for (int e = tid; e < 16*SUH; e += 256) sacc[e] = 0.f;
    // stage previous outputs v for this batch tile (padded, conflict-free reads)
    if (iter > 0) {
        for (int e = tid; e < 16*JD_; e += 256) {
            int b = e >> 10, jd = e & (JD_-1);
            vls[b*SV + (jd >> 5)*SVJ + (jd & 31)] = vprev[(size_t)(b0+b)*JD_ + jd];
        }
    }
    __syncthreads();

    for (int ic = 0; ic < ICHUNK; ++ic) {
        const int i = i0 + ic;

        // prefetch a slice of next i's W into cache (W is L2-resident after pass 1)
        if (ic + 1 < ICHUNK) {
            const float* pf = W + (size_t)(wave*4)*(IN_*D_*DIN_)
                                + (size_t)(i+1)*(D_*DIN_) + lane*16;
            __builtin_prefetch(pf, 0, 0);
        }

        // ---------- phase 1: u_hat[16b x 1024jd] via V_WMMA_F32_16X16X4_F32 ----------
        // B operand (K x N = k x batch): lane holds B[k = r + 2*hi][n = lo] in vgpr r
        v2f bf[4];
        {
            const float* ib = inp + (size_t)(b0 + lo)*(IN_*DIN_) + (size_t)i*DIN_ + 2*hi;
#pragma unroll
            for (int s4 = 0; s4 < 4; ++s4) bf[s4] = *(const v2f*)(ib + 4*s4);
        }

        // each wave computes 8 jd-tiles of 16 rows
#pragma unroll
        for (int tt8 = 0; tt8 < 8; ++tt8) {
            const int tt    = wave*8 + tt8;
            const int jdrow = tt*16 + lo;                 // A: lane holds M = lo
            const int j = jdrow >> 5, d = jdrow & 31;
            const float* wb = W + (size_t)j*(IN_*D_*DIN_) + (size_t)i*(D_*DIN_)
                                + d*DIN_ + 2*hi;          // A: k = 4*s4 + 2*hi + r
            v8f acc = {};
#pragma unroll
            for (int s4 = 0; s4 < 4; ++s4) {
                v2f af = *(const v2f*)(wb + 4*s4);
                acc = __builtin_amdgcn_wmma_f32_16x16x4_f32(
                        false, af, false, bf[s4], (short)0, acc, false, false);
            }
            // D layout: lane holds D[M = r + 8*hi][N = lo] -> uhat[b=lo][jd]
            float* dst = &uhat[lo*SUH + tt*16 + 8*hi];
            *(float4*)dst       = make_float4(acc[0], acc[1], acc[2], acc[3]);
            *((float4*)dst + 1) = make_float4(acc[4], acc[5], acc[6], acc[7]);
        }
        __syncthreads();

        // ---------- phase 2: routing math for this i ----------
        if (iter > 0) {
            // agreement a[b,j] = sum_d u_hat[b,jd]*v[b,j,d]; update logits
#pragma unroll
            for (int pp = 0; pp < 2; ++pp) {
                int p = tid + pp*256;         // 512 = 16b x 32j
                int b = p & 15, j = p >> 4;   // b fast within wave -> conflict-free LDS
                float a = 0.f;
                const float* ur = &uhat[b*SUH + j*32];
                const float* vr = &vls [b*SV  + j*SVJ];
#pragma unroll
                for (int d = 0; d < 32; ++d) a += ur[d] * vr[d];
                size_t bi = (size_t)(b0+b)*(J_*IN_) + (size_t)j*IN_ + i;
                float bl = blog[bi] + a;
                blog[bi] = bl;                // persists for next iteration
                cls[b*32 + j] = bl;
            }
            __syncthreads();
            // softmax over j per (b,i)
            if (tid < 16) {
                float m = -1e30f;
                for (int j = 0; j < 32; ++j) m = fmaxf(m, cls[tid*32 + j]);
                float se = 0.f;
                for (int j = 0; j < 32; ++j) se += expf(cls[tid*32 + j] - m);
                float inv = 1.f / se;
                for (int j = 0; j < 32; ++j)
                    cls[tid*32 + j] = expf(cls[tid*32 + j] - m) * inv;
            }
        } else {
            // iter 0: softmax of zeros -> uniform 1/J
            for (int p = tid; p < 512; p += 256) cls[p] = 1.f / 32.f;
        }
        __syncthreads();

        // s[b,jd] += c[b,j] * u_hat[b,jd]
        for (int e = tid; e < 16*JD_; e += 256) {
            int b = e >> 10, jd = e & (JD_-1);
            sacc[b*SUH + jd] += cls[b*32 + (jd >> 5)] * uhat[b*SUH + jd];
        }
        __syncthreads();   // protect uhat before next i's WMMA writes
    }

    // hierarchical reduction: one f32 atomic per element per workgroup
    for (int e = tid; e < 16*JD_; e += 256) {
        int b = e >> 10, jd = e & (JD_-1);
        atomicAdd(&sout[(size_t)(b0+b)*JD_ + jd], sacc[b*SUH + jd]);
    }
}

// squash: one wave32 handles one (b,j) row of D=32 elements
__global__ __launch_bounds__(256, 1)
void caps_squash(const float* __restrict__ s, float* __restrict__ out) {
    int idx = blockIdx.x * 256 + threadIdx.x;   // 65536 total
    float val = s[idx];
    float s2 = val * val;
#pragma unroll
    for (int m = 16; m > 0; m >>= 1) s2 += __shfl_xor(s2, m, 32);
    float scale = (s2 / (1.f + s2)) / sqrtf(s2 + 1e-7f);
    out[idx] = val * scale;
}

extern "C" void kernel_launch(void* const* d_in, const int* in_sizes, int n_in,
                              void* d_out, int out_size, void* d_ws, size_t ws_size,
                              hipStream_t stream) {
    const float* inputs = (const float*)d_in[0];  // [64, 2048, 16]
    const float* W      = (const float*)d_in[1];  // [32, 2048, 32, 16]

    float* ws   = (float*)d_ws;
    float* blog = ws;                         // 64*32*2048 = 4,194,304 floats
    float* s    = blog + (size_t)B_*J_*IN_;   // 65536 floats
    float* v    = s + B_*JD_;                 // 65536 floats

    // logits start at zero each call (deterministic across graph replays)
    hipMemsetAsync(blog, 0, (size_t)B_*J_*IN_*sizeof(float), stream);

    const size_t lds = (size_t)LDS_FLOATS * sizeof(float);
    dim3 grid(IN_/ICHUNK, B_/16, 1);          // 128 x 4 workgroups

    for (int it = 0; it < 3; ++it) {
        hipMemsetAsync(s, 0, (size_t)B_*JD_*sizeof(float), stream);
        caps_route<<<grid, 256, lds, stream>>>(inputs, W, v, blog, s, it);
        float* outp = (it == 2) ? (float*)d_out : v;
        caps_squash<<<B_*JD_/256, 256, 0, stream>>>(s, outp);
    }
}